// STGraphNet_41308995453090
// MI455X (gfx1250) — compile-verified
//
#include <hip/hip_runtime.h>
#include <math.h>
#include <stdint.h>

namespace {

constexpr int kN    = 16384;
constexpr int kWin  = 64;
constexpr int kB    = 256;      // graphs
constexpr int kE    = 262144;
constexpr int kEtot = kE + kN;  // + self loops
constexpr int kHid  = 256;
constexpr int kDh   = 32;       // per-head dim
constexpr int kFreq = 33;
constexpr float kEps = 1e-5f;

typedef __attribute__((ext_vector_type(2))) float v2f;
typedef __attribute__((ext_vector_type(8))) float v8f;

__device__ __forceinline__ float gelu_f(float v) {
  return 0.5f * v * (1.0f + erff(v * 0.7071067811865476f));
}
// monotonic int encoding of float for atomicMax-based segment max
__device__ __forceinline__ int ford(float f) {
  int i = __float_as_int(f);
  return i >= 0 ? i : (i ^ 0x7FFFFFFF);
}
__device__ __forceinline__ float ordf(int o) {
  return __int_as_float(o >= 0 ? o : (o ^ 0x7FFFFFFF));
}

// CDNA5 async copy: global -> LDS, 16B per issue, tracked by ASYNCcnt.
// VDST = VGPR with LDS byte offset, ADDR = 64-bit global address (GV mode).
__device__ __forceinline__ void async_copy_b128(uint32_t lds_byte_off, const void* gptr) {
  asm volatile("global_load_async_to_lds_b128 %0, %1, off"
               :: "v"(lds_byte_off), "v"((uint64_t)(uintptr_t)gptr)
               : "memory");
}
__device__ __forceinline__ void wait_async0() {
#if __has_builtin(__builtin_amdgcn_s_wait_asynccnt)
  __builtin_amdgcn_s_wait_asynccnt(0);
#else
  asm volatile("s_wait_asynccnt 0" ::: "memory");
#endif
}

__global__ void fill_f32(float* __restrict__ p, float v, int n) {
  for (int i = blockIdx.x * blockDim.x + threadIdx.x; i < n; i += gridDim.x * blockDim.x) p[i] = v;
}
__global__ void fill_i32(int* __restrict__ p, int v, int n) {
  for (int i = blockIdx.x * blockDim.x + threadIdx.x; i < n; i += gridDim.x * blockDim.x) p[i] = v;
}

// h = gelu(x @ enc_W + enc_b), x:[N,4] W:[4,256]
__global__ void encoder_k(const float* __restrict__ x, const float* __restrict__ W,
                          const float* __restrict__ b, float* __restrict__ h) {
  int n = blockIdx.x, j = threadIdx.x;
  float acc = b[j];
#pragma unroll
  for (int f = 0; f < 4; ++f) acc += x[n * 4 + f] * W[f * kHid + j];
  h[(size_t)n * kHid + j] = gelu_f(acc);
}

// ---------------------------------------------------------------------------
// C[M,256] = A[M,256] @ W[256,256] via V_WMMA_F32_16X16X4_F32 with W staged in
// LDS (256 KB of the WGP's 320 KB) through global_load_async_to_lds_b128.
// Block = 8 waves; each wave owns a 16-row strip and sweeps 16 n-tiles, pulling
// B-fragments from LDS (ds_load) and A-fragments from global (L0-resident).
// ISA frag layouts (CDNA5 ISA §7.12.2):
//   A 16x4:  lanes 0-15 -> K={kb,kb+1}; lanes 16-31 -> K={kb+2,kb+3}
//   C 16x16: VGPR r -> M = r (lanes 0-15) / r+8 (lanes 16-31), N = lane&15
// ---------------------------------------------------------------------------
__global__ void gemm_wmma_lds(const float* __restrict__ A, const float* __restrict__ W,
                              float* __restrict__ C, int M, int lda, int ldc) {
  extern __shared__ float lw[];  // 256*256 floats
  {
    int tid = threadIdx.x;
    // 16384 float4 chunks, 64 per thread, coalesced across the block
#pragma unroll 4
    for (int i = 0; i < 64; ++i) {
      int fidx = (i * 256 + tid) * 4;  // float index
      async_copy_b128((uint32_t)(uintptr_t)(&lw[fidx]), W + fidx);
    }
  }
  wait_async0();       // this wave's async copies landed in LDS
  __syncthreads();     // all waves' copies visible block-wide

  int wave = threadIdx.x >> 5;
  int lane = threadIdx.x & 31;
  int half = lane >> 4;
  int l15  = lane & 15;
  int mbase = blockIdx.x * 128 + wave * 16;
  if (mbase >= M) return;  // wave-uniform

  const float* arow = A + (size_t)(mbase + l15) * lda + half * 2;
  for (int nt = 0; nt < 16; ++nt) {
    int col = nt * 16 + l15;
    const float* lc = lw + (size_t)(half * 2) * kHid + col;
    v8f acc0 = {}; v8f acc1 = {};
    for (int k = 0; k < kHid; k += 8) {
      v2f a0, b0, a1, b1;
      a0.x = arow[k];     a0.y = arow[k + 1];
      b0.x = lc[k * kHid];       b0.y = lc[(k + 1) * kHid];
      a1.x = arow[k + 4]; a1.y = arow[k + 5];
      b1.x = lc[(k + 4) * kHid]; b1.y = lc[(k + 5) * kHid];
      acc0 = __builtin_amdgcn_wmma_f32_16x16x4_f32(false, a0, false, b0, (short)0, acc0, false, false);
      acc1 = __builtin_amdgcn_wmma_f32_16x16x4_f32(false, a1, false, b1, (short)0, acc1, false, false);
    }
    v8f acc = acc0 + acc1;
    int row = mbase + half * 8;
#pragma unroll
    for (int r = 0; r < 8; ++r) C[(size_t)(row + r) * ldc + col] = acc[r];
  }
}

// Generic global-operand WMMA GEMM (used for the K=512 head GEMM).
__global__ void gemm_wmma_f32(const float* __restrict__ A, const float* __restrict__ W,
                              float* __restrict__ C, int M, int K, int N, int lda, int ldc) {
  int wave = (int)((blockIdx.x * blockDim.x + threadIdx.x) >> 5);
  int lane = threadIdx.x & 31;
  int ntiles = N >> 4;
  int mt = wave / ntiles;
  int nt = wave - mt * ntiles;
  if (mt * 16 >= M) return;  // wave-uniform
  int half = lane >> 4;
  int l15  = lane & 15;
  const float* arow = A + (size_t)(mt * 16 + l15) * lda + half * 2;
  const float* wcol = W + (size_t)(half * 2) * N + (nt * 16 + l15);
  v8f acc0 = {}; v8f acc1 = {};
  for (int k = 0; k < K; k += 8) {
    v2f a0, b0, a1, b1;
    a0.x = arow[k];     a0.y = arow[k + 1];
    b0.x = wcol[(size_t)k * N];       b0.y = wcol[(size_t)(k + 1) * N];
    a1.x = arow[k + 4]; a1.y = arow[k + 5];
    b1.x = wcol[(size_t)(k + 4) * N]; b1.y = wcol[(size_t)(k + 5) * N];
    acc0 = __builtin_amdgcn_wmma_f32_16x16x4_f32(false, a0, false, b0, (short)0, acc0, false, false);
    acc1 = __builtin_amdgcn_wmma_f32_16x16x4_f32(false, a1, false, b1, (short)0, acc1, false, false);
  }
  v8f acc = acc0 + acc1;
  int col = nt * 16 + l15;
  int row = mt * 16 + half * 8;
#pragma unroll
  for (int r = 0; r < 8; ++r) C[(size_t)(row + r) * ldc + col] = acc[r];
}

// per (node, head) attention logits: a = sum_d hh[n,h,d] * att[h,d]
__global__ void att_k(const float* __restrict__ X, const float* __restrict__ att_s,
                      const float* __restrict__ att_d, float* __restrict__ as,
                      float* __restrict__ ad) {
  int idx = blockIdx.x * blockDim.x + threadIdx.x;  // n*8 + h
  int n = idx >> 3, h = idx & 7;
  const float* row = X + (size_t)n * kHid + h * kDh;
  float s = 0.f, d = 0.f;
#pragma unroll
  for (int t = 0; t < kDh; ++t) {
    s += row[t] * att_s[h * kDh + t];
    d += row[t] * att_d[h * kDh + t];
  }
  as[idx] = s; ad[idx] = d;
}

__device__ __forceinline__ void edge_nodes(const int* ei, int e, int& s, int& d) {
  if (e < kE) { s = ei[e]; d = ei[kE + e]; } else { s = d = e - kE; }
}

__global__ void gat_e1(const int* __restrict__ ei, const float* __restrict__ as,
                       const float* __restrict__ ad, float* __restrict__ ebuf,
                       int* __restrict__ mI) {
  int idx = blockIdx.x * blockDim.x + threadIdx.x;  // e*8 + h
  int e = idx >> 3, h = idx & 7;
  int s, d; edge_nodes(ei, e, s, d);
  float v = as[s * 8 + h] + ad[d * 8 + h];
  v = v > 0.f ? v : 0.2f * v;  // leaky_relu(., 0.2)
  ebuf[idx] = v;
  atomicMax(&mI[d * 8 + h], ford(v));
}

__global__ void gat_e2(const int* __restrict__ ei, float* __restrict__ ebuf,
                       const int* __restrict__ mI, float* __restrict__ sS) {
  int idx = blockIdx.x * blockDim.x + threadIdx.x;
  int e = idx >> 3, h = idx & 7;
  int s, d; edge_nodes(ei, e, s, d);
  float ex = expf(ebuf[idx] - ordf(mI[d * 8 + h]));
  ebuf[idx] = ex;
  atomicAdd(&sS[d * 8 + h], ex);
}

// O[dst,h,:] += hh[src,h,:] * alpha ; one block (8 heads x 32 dims) per edge
__global__ void gat_scatter(const int* __restrict__ ei, const float* __restrict__ ebuf,
                            const float* __restrict__ sS, const float* __restrict__ X,
                            float* __restrict__ O) {
  int e = blockIdx.x;
  int h = threadIdx.x >> 5, t = threadIdx.x & 31;
  int s, d; edge_nodes(ei, e, s, d);
  float alpha = ebuf[e * 8 + h] / (sS[d * 8 + h] + 1e-16f);
  atomicAdd(&O[(size_t)d * kHid + h * kDh + t], X[(size_t)s * kHid + h * kDh + t] * alpha);
}

// fused (optional bias) + LayerNorm + (optional exact GELU); one block per node row
__global__ void ln_k(const float* __restrict__ in, const float* __restrict__ bias,
                     const float* __restrict__ g, const float* __restrict__ b,
                     float* __restrict__ out, int do_gelu) {
  __shared__ float red[kHid];
  int n = blockIdx.x, j = threadIdx.x;
  float v = in[(size_t)n * kHid + j] + (bias ? bias[j] : 0.f);
  red[j] = v; __syncthreads();
  for (int s = kHid / 2; s > 0; s >>= 1) { if (j < s) red[j] += red[j + s]; __syncthreads(); }
  float mu = red[0] * (1.f / kHid); __syncthreads();
  float c = v - mu;
  red[j] = c * c; __syncthreads();
  for (int s = kHid / 2; s > 0; s >>= 1) { if (j < s) red[j] += red[j + s]; __syncthreads(); }
  float var = red[0] * (1.f / kHid);
  float y = c * rsqrtf(var + kEps) * g[j] + b[j];
  if (do_gelu) y = gelu_f(y);
  out[(size_t)n * kHid + j] = y;
}

__global__ void deg_k(const int* __restrict__ ei, const float* __restrict__ ea,
                      float* __restrict__ deg) {
  int e = blockIdx.x * blockDim.x + threadIdx.x;
  int s, d; edge_nodes(ei, e, s, d);
  float w = (e < kE) ? ea[e] : 1.f;
  atomicAdd(&deg[d], w);
}

__global__ void norm_k(const int* __restrict__ ei, const float* __restrict__ ea,
                       const float* __restrict__ deg, float* __restrict__ nrm) {
  int e = blockIdx.x * blockDim.x + threadIdx.x;
  int s, d; edge_nodes(ei, e, s, d);
  float w  = (e < kE) ? ea[e] : 1.f;
  float ds = deg[s], dd = deg[d];
  float is = ds > 0.f ? rsqrtf(fmaxf(ds, 1e-12f)) : 0.f;
  float id = dd > 0.f ? rsqrtf(fmaxf(dd, 1e-12f)) : 0.f;
  nrm[e] = is * w * id;
}

// O[dst,:] += X[src,:] * norm[e] ; one 256-thread block per edge
__global__ void gcn_scatter(const int* __restrict__ ei, const float* __restrict__ nrm,
                            const float* __restrict__ X, float* __restrict__ O) {
  int e = blockIdx.x, j = threadIdx.x;
  int s, d; edge_nodes(ei, e, s, d);
  atomicAdd(&O[(size_t)d * kHid + j], X[(size_t)s * kHid + j] * nrm[e]);
}

__global__ void add_k(float* __restrict__ p, const float* __restrict__ r, int n) {
  int i = blockIdx.x * blockDim.x + threadIdx.x;
  if (i < n) p[i] += r[i];
}

__global__ void pool_k(const float* __restrict__ h, float* __restrict__ pooled) {
  int g = blockIdx.x, j = threadIdx.x;
  float s = 0.f;
  for (int t = 0; t < kWin; ++t) s += h[(size_t)(g * kWin + t) * kHid + j];
  pooled[(size_t)g * kHid + j] = s * (1.f / kWin);
}

__global__ void biasgelu_k(float* __restrict__ p, const float* __restrict__ b, int ld) {
  int r = blockIdx.x, c = threadIdx.x;
  p[(size_t)r * ld + c] = gelu_f(p[(size_t)r * ld + c] + b[c]);
}

// |rfft(ts)| per graph; ts = x[:,0] windows of 64
__global__ void dft_k(const float* __restrict__ x, float* __restrict__ mag) {
  __shared__ float ts[kWin];
  int g = blockIdx.x, t = threadIdx.x;
  ts[t] = x[(g * kWin + t) * 4 + 0];
  __syncthreads();
  if (t < kFreq) {
    float re = 0.f, im = 0.f;
    for (int u = 0; u < kWin; ++u) {
      float ang = -6.283185307179586f * (float)(t * u) * (1.f / kWin);
      re += ts[u] * cosf(ang);
      im += ts[u] * sinf(ang);
    }
    mag[g * kFreq + t] = sqrtf(re * re + im * im);
  }
}

// freq_feat = gelu(mag @ fq_W + fq_b) -> comb[:, 256:512]  (K=33, scalar)
__global__ void fq_k(const float* __restrict__ mag, const float* __restrict__ W,
                     const float* __restrict__ b, float* __restrict__ comb) {
  int g = blockIdx.x, j = threadIdx.x;
  float acc = b[j];
  for (int f = 0; f < kFreq; ++f) acc += mag[g * kFreq + f] * W[f * kHid + j];
  comb[(size_t)g * 512 + kHid + j] = gelu_f(acc);
}

// out[g] = h1[g,:] . h2_W[:,0] + h2_b
__global__ void head_k(const float* __restrict__ h1, const float* __restrict__ w,
                       const float* __restrict__ b, float* __restrict__ out) {
  __shared__ float red[kHid];
  int g = blockIdx.x, j = threadIdx.x;
  red[j] = h1[(size_t)g * kHid + j] * w[j];
  __syncthreads();
  for (int s = kHid / 2; s > 0; s >>= 1) { if (j < s) red[j] += red[j + s]; __syncthreads(); }
  if (j == 0) out[g] = red[0] + b[0];
}

}  // namespace

extern "C" void kernel_launch(void* const* d_in, const int* in_sizes, int n_in,
                              void* d_out, int out_size, void* d_ws, size_t ws_size,
                              hipStream_t stream) {
  (void)in_sizes; (void)n_in; (void)out_size; (void)ws_size;
  const float* x        = (const float*)d_in[0];
  const int*   ei       = (const int*)d_in[1];   // [2,E]
  const float* ea       = (const float*)d_in[2]; // [E,1]
  /* d_in[3] = batch (derivable as n/64, unused) */
  const float* enc_W    = (const float*)d_in[4];
  const float* enc_b    = (const float*)d_in[5];
  const float* gat_W    = (const float*)d_in[6];
  const float* att_s    = (const float*)d_in[7];
  const float* att_d    = (const float*)d_in[8];
  const float* gat_bias = (const float*)d_in[9];
  const float* gat_g    = (const float*)d_in[10];
  const float* gat_b    = (const float*)d_in[11];
  const float* gcn_W    = (const float*)d_in[12]; // [8,256,256]
  const float* gcn_b    = (const float*)d_in[13];
  const float* gcn_g    = (const float*)d_in[14];
  const float* gcn_bb   = (const float*)d_in[15];
  const float* ro_W     = (const float*)d_in[16];
  const float* ro_b     = (const float*)d_in[17];
  const float* fq_W     = (const float*)d_in[18];
  const float* fq_b     = (const float*)d_in[19];
  const float* h1_W     = (const float*)d_in[20];
  const float* h1_b     = (const float*)d_in[21];
  const float* h2_W     = (const float*)d_in[22];
  const float* h2_b     = (const float*)d_in[23];
  float* out = (float*)d_out;

  // workspace layout (~80 MB)
  float* ws = (float*)d_ws;
  size_t NB = (size_t)kN * kHid;
  float* P      = ws;             // current node features
  float* Q      = P + NB;         // scatter target / intermediate
  float* R      = Q + NB;         // second scatter target
  float* X      = R + NB;         // GEMM output (xw / hh)
  float* deg    = X + NB;         // [N]
  float* nrm    = deg + kN;       // [Etot]
  float* asv    = nrm + kEtot;    // [N,8]
  float* adv    = asv + (size_t)kN * 8;
  int*   mI     = (int*)(adv + (size_t)kN * 8);   // [N,8] ordered-int max
  float* sS     = (float*)(mI + (size_t)kN * 8);  // [N,8] softmax denom
  float* ebuf   = sS + (size_t)kN * 8;            // [Etot,8]
  float* pooled = ebuf + (size_t)kEtot * 8;       // [B,256]
  float* comb   = pooled + (size_t)kB * kHid;     // [B,512]
  float* mag    = comb + (size_t)kB * 512;        // [B,33]
  float* h1o    = mag + (size_t)kB * kFreq;       // [B,256]

  dim3 b256(256);
  constexpr size_t kWLds = (size_t)kHid * kHid * sizeof(float);  // 256 KB
  // C[M,256] = A[M,256] @ W[256,256], W staged via async-to-LDS
  auto gemm256 = [&](const float* A, const float* W, float* C, int M, int lda, int ldc) {
    gemm_wmma_lds<<<dim3((M + 127) / 128), b256, kWLds, stream>>>(A, W, C, M, lda, ldc);
  };

  // ---- encoder ----
  encoder_k<<<dim3(kN), b256, 0, stream>>>(x, enc_W, enc_b, P);

  // ---- GAT block + LN ----
  gemm256(P, gat_W, X, kN, kHid, kHid);
  att_k<<<dim3(kN * 8 / 256), b256, 0, stream>>>(X, att_s, att_d, asv, adv);
  fill_i32<<<dim3(512), b256, 0, stream>>>(mI, (int)0x807FFFFF /* ford(-inf) */, kN * 8);
  fill_f32<<<dim3(512), b256, 0, stream>>>(sS, 0.f, kN * 8);
  fill_f32<<<dim3(4096), b256, 0, stream>>>(Q, 0.f, (int)NB);
  gat_e1<<<dim3(kEtot * 8 / 256), b256, 0, stream>>>(ei, asv, adv, ebuf, mI);
  gat_e2<<<dim3(kEtot * 8 / 256), b256, 0, stream>>>(ei, ebuf, mI, sS);
  gat_scatter<<<dim3(kEtot), b256, 0, stream>>>(ei, ebuf, sS, X, Q);
  ln_k<<<dim3(kN), b256, 0, stream>>>(Q, gat_bias, gat_g, gat_b, P, 0);

  // ---- GCN normalization (once) ----
  fill_f32<<<dim3(64), b256, 0, stream>>>(deg, 0.f, kN);
  deg_k<<<dim3(kEtot / 256), b256, 0, stream>>>(ei, ea, deg);
  norm_k<<<dim3(kEtot / 256), b256, 0, stream>>>(ei, ea, deg, nrm);

  // ---- 4 residual blocks of 2 GCN layers ----
  for (int blk = 0; blk < 4; ++blk) {
    int l0 = 2 * blk, l1 = 2 * blk + 1;
    gemm256(P, gcn_W + (size_t)l0 * kHid * kHid, X, kN, kHid, kHid);
    fill_f32<<<dim3(4096), b256, 0, stream>>>(Q, 0.f, (int)NB);
    gcn_scatter<<<dim3(kEtot), b256, 0, stream>>>(ei, nrm, X, Q);
    ln_k<<<dim3(kN), b256, 0, stream>>>(Q, gcn_b + l0 * kHid, gcn_g + l0 * kHid,
                                        gcn_bb + l0 * kHid, Q, 1);
    gemm256(Q, gcn_W + (size_t)l1 * kHid * kHid, X, kN, kHid, kHid);
    fill_f32<<<dim3(4096), b256, 0, stream>>>(R, 0.f, (int)NB);
    gcn_scatter<<<dim3(kEtot), b256, 0, stream>>>(ei, nrm, X, R);
    ln_k<<<dim3(kN), b256, 0, stream>>>(R, gcn_b + l1 * kHid, gcn_g + l1 * kHid,
                                        gcn_bb + l1 * kHid, R, 1);
    add_k<<<dim3((int)(NB / 256)), b256, 0, stream>>>(P, R, (int)NB);  // residual
  }

  // ---- readout ----
  pool_k<<<dim3(kB), b256, 0, stream>>>(P, pooled);
  gemm256(pooled, ro_W, comb, kB, kHid, 512);                   // -> comb[:, :256]
  biasgelu_k<<<dim3(kB), b256, 0, stream>>>(comb, ro_b, 512);
  dft_k<<<dim3(kB), dim3(kWin), 0, stream>>>(x, mag);
  fq_k<<<dim3(kB), b256, 0, stream>>>(mag, fq_W, fq_b, comb);   // -> comb[:, 256:]
  {
    int waves = (kB / 16) * (kHid / 16);  // 256 waves -> 32 blocks
    gemm_wmma_f32<<<dim3(waves / 8), b256, 0, stream>>>(comb, h1_W, h1o,
                                                        kB, 512, kHid, 512, kHid);
  }
  biasgelu_k<<<dim3(kB), b256, 0, stream>>>(h1o, h1_b, kHid);
  head_k<<<dim3(kB), b256, 0, stream>>>(h1o, h2_W, h2_b, out);
}